// EmbeddedGCN_57062935495223
// MI455X (gfx1250) — compile-verified
//
#include <hip/hip_runtime.h>

// Shapes (fixed by the reference)
#define BATCH   16
#define M_DIM   128
#define FEACH   64
#define FIN     128   // concat features = K of the weight GEMM
#define FOUT    128
#define PSTRIDE 132   // padded LDS row stride (floats) -> conflict-free b64 reads

typedef float v2f __attribute__((ext_vector_type(2)));
typedef float v8f __attribute__((ext_vector_type(8)));

#define FMA4(av, fv)                          \
    acc.x = fmaf((av), (fv).x, acc.x);        \
    acc.y = fmaf((av), (fv).y, acc.y);        \
    acc.z = fmaf((av), (fv).z, acc.z);        \
    acc.w = fmaf((av), (fv).w, acc.w)

__global__ __launch_bounds__(256)
void gcn_pool_gemm_fused(const float* __restrict__ s1,
                         const float* __restrict__ s2,
                         const float* __restrict__ adj,
                         const float* __restrict__ W,     // (FIN, FOUT) row-major
                         const float* __restrict__ bias,  // (FOUT)
                         float* __restrict__ out)         // (B, M, FOUT)
{
    __shared__ float pooled[16 * PSTRIDE];

    const int tid    = threadIdx.x;
    const int lane   = tid & 31;
    const int wave   = tid >> 5;
    const int bx     = blockIdx.x;
    const int b      = bx >> 3;          // batch
    const int i_base = (bx & 7) << 4;    // 16-row tile of i

    const int hi = lane >> 4;            // 0: features 0..63 (s1), 1: 64..127 (s2)
    const int l4 = (lane & 15) << 2;     // 4 features per lane within the half

    // ---------------- Phase 1: masked sum-pool over j (HBM-bound streaming) ----
    // pooled[il][f] = sum_j adj[b, i_base+il, j] * concat(s1,s2)[b, i_base+il, j, f]
    for (int rr = 0; rr < 2; ++rr) {
        const int il = wave * 2 + rr;                 // local row 0..15
        const int i  = i_base + il;
        const size_t fb   = ((size_t)(b * M_DIM + i)) * M_DIM * FEACH;
        const float* fptr = (hi ? s2 : s1) + fb + l4; // lane-split across the two tensors
        const float* arow = adj + (size_t)(b * M_DIM + i) * M_DIM;

        float4 acc = {0.f, 0.f, 0.f, 0.f};
        #pragma unroll 2
        for (int j4 = 0; j4 < M_DIM; j4 += 4) {
            const float4 a4 = *(const float4*)(arow + j4);           // broadcast load
            const float4 f0 = *(const float4*)(fptr + (size_t)(j4 + 0) * FEACH);
            const float4 f1 = *(const float4*)(fptr + (size_t)(j4 + 1) * FEACH);
            const float4 f2 = *(const float4*)(fptr + (size_t)(j4 + 2) * FEACH);
            const float4 f3 = *(const float4*)(fptr + (size_t)(j4 + 3) * FEACH);
            FMA4(a4.x, f0);
            FMA4(a4.y, f1);
            FMA4(a4.z, f2);
            FMA4(a4.w, f3);
        }
        *(float4*)&pooled[il * PSTRIDE + hi * FEACH + l4] = acc;  // 16B aligned
    }
    __syncthreads();

    // ---------------- Phase 2: out_tile = relu(pooled(16xK) @ W(KxN) + bias) ----
    // One 16x16 output column tile per wave, fp32 WMMA, K stepped by 4.
    const int n0 = wave << 4;            // column tile base
    const int m  = lane & 15;

    v8f cacc = {0.f, 0.f, 0.f, 0.f, 0.f, 0.f, 0.f, 0.f};
    #pragma unroll 4
    for (int k0 = 0; k0 < FIN; k0 += 4) {
        const int kk = k0 + 2 * hi;      // lanes 0-15: K={k0,k0+1}; 16-31: K={k0+2,k0+3}
        // A fragment (16x4 f32): row m, two consecutive K values per lane
        const v2f afrag = *(const v2f*)&pooled[m * PSTRIDE + kk];
        // B fragment (4x16 f32): rows K=kk,kk+1, column n0+m (coalesced within rows)
        v2f bfrag;
        bfrag.x = W[(size_t)kk * FOUT + n0 + m];
        bfrag.y = W[(size_t)(kk + 1) * FOUT + n0 + m];
        cacc = __builtin_amdgcn_wmma_f32_16x16x4_f32(
            /*neg_a=*/false, afrag, /*neg_b=*/false, bfrag,
            /*c_mod=*/(short)0, cacc, /*reuse_a=*/false, /*reuse_b=*/false);
    }

    // Epilogue: bias + ReLU + store. C/D layout: VGPR r -> M = r (lanes 0-15),
    // M = r+8 (lanes 16-31); N = lane % 16.
    const float bv = bias[n0 + m];
    float* obase = out + (size_t)b * M_DIM * FOUT;
    #pragma unroll
    for (int r = 0; r < 8; ++r) {
        const int i = i_base + r + hi * 8;
        float v = cacc[r] + bv;
        v = v > 0.f ? v : 0.f;
        obase[(size_t)i * FOUT + n0 + m] = v;
    }
}

extern "C" void kernel_launch(void* const* d_in, const int* in_sizes, int n_in,
                              void* d_out, int out_size, void* d_ws, size_t ws_size,
                              hipStream_t stream) {
    const float* s1   = (const float*)d_in[0];  // (16,128,128,64)
    const float* s2   = (const float*)d_in[1];  // (16,128,128,64)
    const float* adj  = (const float*)d_in[2];  // (16,128,128)
    const float* W    = (const float*)d_in[3];  // (128,128)
    const float* bias = (const float*)d_in[4];  // (128)
    float* out        = (float*)d_out;          // (16,128,128)

    dim3 grid(BATCH * (M_DIM / 16));  // 128 workgroups
    dim3 block(256);                  // 8 waves (wave32)
    gcn_pool_gemm_fused<<<grid, block, 0, stream>>>(s1, s2, adj, W, bias, out);
}